// PMACMultiAspect_41979010351486
// MI455X (gfx1250) — compile-verified
//
#include <hip/hip_runtime.h>
#include <math.h>

// Problem constants (match reference)
#define Bsz  2048
#define Asz  16
#define Dsz  768
#define Hsz  128
#define H2sz 64
#define Lsz  2
#define ROWS 16      // batch rows per workgroup (one WMMA M-tile)
#define TPB  256     // 8 wave32s

// ---- CDNA5 async global->LDS support (guarded; falls back to sync copy) ----
#if defined(__has_builtin)
#  if __has_builtin(__builtin_amdgcn_global_load_async_to_lds_b128)
#    define HAVE_ASYNC_LDS 1
#  endif
#  if __has_builtin(__builtin_amdgcn_s_wait_asynccnt)
#    define HAVE_WAIT_ASYNC 1
#  endif
#endif

typedef float v2f __attribute__((ext_vector_type(2)));
typedef float v8f __attribute__((ext_vector_type(8)));

#if defined(HAVE_ASYNC_LDS)
typedef int v4i __attribute__((ext_vector_type(4)));
typedef __attribute__((address_space(1))) v4i* gv4i_p;   // global (AS1)
typedef __attribute__((address_space(3))) v4i* lv4i_p;   // LDS (AS3)
#endif

static __device__ __forceinline__ v8f vzero8() {
  v8f z;
#pragma unroll
  for (int i = 0; i < 8; ++i) z[i] = 0.0f;
  return z;
}

// Full-precision matrix core op: D = A(16x4 f32) * B(4x16 f32) + C(16x16 f32)
static __device__ __forceinline__ v8f wmma4(v2f a, v2f b, v8f c) {
  return __builtin_amdgcn_wmma_f32_16x16x4_f32(false, a, false, b, (short)0, c,
                                               false, false);
}

// Wait for all outstanding async global->LDS transfers of this wave.
static __device__ __forceinline__ void wait_async_all() {
#if defined(HAVE_ASYNC_LDS)
#  if defined(HAVE_WAIT_ASYNC)
  __builtin_amdgcn_s_wait_asynccnt(0);
#  else
  asm volatile("s_wait_asynccnt 0x0" ::: "memory");
#  endif
#endif
}

// Prefetch aspect `a` of this block's 16 rows into LDS buffer `dst`
// (16 x 768 f32 = 3072 async B128 transfers, ASYNCcnt-tracked).
static __device__ __forceinline__ void prefetch_f2(const float* __restrict__ features,
                                                   int b0, int a, float* dst, int tid) {
#if defined(HAVE_ASYNC_LDS)
  for (int v = tid; v < ROWS * (Dsz / 4); v += TPB) {
    int r = v / (Dsz / 4);
    int c4 = v % (Dsz / 4);
    const float* g = features + ((size_t)(b0 + r) * Asz + a) * Dsz + c4 * 4;
    float* l = dst + r * Dsz + c4 * 4;
    __builtin_amdgcn_global_load_async_to_lds_b128((gv4i_p)g, (lv4i_p)l, 0, 0);
  }
#else
  for (int idx = tid; idx < ROWS * Dsz; idx += TPB) {
    int r = idx / Dsz, c = idx % Dsz;
    dst[idx] = features[((size_t)(b0 + r) * Asz + a) * Dsz + c];
  }
#endif
}

// One 16x16 output tile of  A[16xK] (LDS, row-major, lda) @ W[KxN] (global, ldw)
// lane roles per ISA layout: m = lane&15 (A row == B col), kh = lane>>4.
static __device__ __forceinline__ v8f gemm_tile(const float* __restrict__ A, int lda,
                                                const float* __restrict__ W, int ldw,
                                                int K, int n0, int m, int kh) {
  v8f acc = vzero8();
  const float* ar = A + m * lda + 2 * kh;
  const float* wr = W + (size_t)(2 * kh) * ldw + n0 + m;
#pragma unroll 4
  for (int k0 = 0; k0 < K; k0 += 4) {
    v2f a, b;
    a.x = ar[k0];
    a.y = ar[k0 + 1];
    b.x = wr[(size_t)k0 * ldw];
    b.y = wr[(size_t)k0 * ldw + ldw];
    acc = wmma4(a, b, acc);
  }
  return acc;
}

// Two A-operands sharing one B-operand stream (k_comp and k_f2 vs same wk tile)
static __device__ __forceinline__ void gemm_tile_dual(
    const float* __restrict__ A1, const float* __restrict__ A2, int lda,
    const float* __restrict__ W, int ldw, int K, int n0, int m, int kh,
    v8f& o1, v8f& o2) {
  v8f acc1 = vzero8(), acc2 = vzero8();
  const float* r1 = A1 + m * lda + 2 * kh;
  const float* r2 = A2 + m * lda + 2 * kh;
  const float* wr = W + (size_t)(2 * kh) * ldw + n0 + m;
#pragma unroll 2
  for (int k0 = 0; k0 < K; k0 += 4) {
    v2f b;
    b.x = wr[(size_t)k0 * ldw];
    b.y = wr[(size_t)k0 * ldw + ldw];
    v2f a1, a2;
    a1.x = r1[k0]; a1.y = r1[k0 + 1];
    a2.x = r2[k0]; a2.y = r2[k0 + 1];
    acc1 = wmma4(a1, b, acc1);
    acc2 = wmma4(a2, b, acc2);
  }
  o1 = acc1;
  o2 = acc2;
}

// A = concat(comp, f2) along K (K = 2*Dsz). Split into two clean halves so the
// A-operand loads stay pure DS (no per-iteration pointer select -> no FLAT).
static __device__ __forceinline__ v8f gemm_cat(const float* __restrict__ C1,
                                               const float* __restrict__ C2,
                                               const float* __restrict__ W, int ldw,
                                               int n0, int m, int kh) {
  v8f acc = vzero8();
  const float* r1 = C1 + m * Dsz + 2 * kh;
  const float* r2 = C2 + m * Dsz + 2 * kh;
  const float* w1 = W + (size_t)(2 * kh) * ldw + n0 + m;
  const float* w2 = W + (size_t)(Dsz + 2 * kh) * ldw + n0 + m;
#pragma unroll 4
  for (int k0 = 0; k0 < Dsz; k0 += 4) {
    v2f a, b;
    a.x = r1[k0];
    a.y = r1[k0 + 1];
    b.x = w1[(size_t)k0 * ldw];
    b.y = w1[(size_t)k0 * ldw + ldw];
    acc = wmma4(a, b, acc);
  }
#pragma unroll 4
  for (int k0 = 0; k0 < Dsz; k0 += 4) {
    v2f a, b;
    a.x = r2[k0];
    a.y = r2[k0 + 1];
    b.x = w2[(size_t)k0 * ldw];
    b.y = w2[(size_t)k0 * ldw + ldw];
    acc = wmma4(a, b, acc);
  }
  return acc;
}

// Scatter a 16x16 D-tile (v8f per lane) into LDS with bias add.
static __device__ __forceinline__ void store_tile(float* __restrict__ O, int ldo,
                                                  v8f acc, int n0, int col16,
                                                  int rbase, const float* __restrict__ bias) {
  float bv = bias[n0 + col16];
#pragma unroll
  for (int r = 0; r < 8; ++r)
    O[(size_t)(rbase + r) * ldo + n0 + col16] = acc[r] + bv;
}

// In-place row LayerNorm over X[16 x N] (+ optional exact GELU), 16 threads/row.
static __device__ void row_ln_act(float* __restrict__ X, int N,
                                  const float* __restrict__ g, const float* __restrict__ b,
                                  float* redS, float* redQ, float* meanS, float* rstdS,
                                  int tid, int do_gelu) {
  int row = tid >> 4, slot = tid & 15;
  float s = 0.f, q = 0.f;
  for (int c = slot; c < N; c += 16) {
    float v = X[row * N + c];
    s += v;
    q += v * v;
  }
  redS[row * 16 + slot] = s;
  redQ[row * 16 + slot] = q;
  __syncthreads();
  if (tid < 16) {
    float ts = 0.f, tq = 0.f;
    for (int i = 0; i < 16; ++i) { ts += redS[tid * 16 + i]; tq += redQ[tid * 16 + i]; }
    float mean = ts / (float)N;
    float var = tq / (float)N - mean * mean;
    meanS[tid] = mean;
    rstdS[tid] = rsqrtf(var + 1e-5f);
  }
  __syncthreads();
  float mean = meanS[row], rr = rstdS[row];
  for (int c = slot; c < N; c += 16) {
    float v = (X[row * N + c] - mean) * rr * g[c] + b[c];
    if (do_gelu) v = 0.5f * v * (1.0f + erff(v * 0.70710678118654752f));
    X[row * N + c] = v;
  }
  __syncthreads();
}

// LDS floats: comp + f2 double buffer + gf + qb (5 * 16*768) + h1 + h2
//           + red (2*256) + stats (6*16) + 32 ints
#define SMEM_FLOATS (5 * ROWS * Dsz + ROWS * Hsz + ROWS * H2sz + 512 + 96 + 32)
#define SMEM_BYTES ((size_t)SMEM_FLOATS * 4)

__global__ __launch_bounds__(TPB, 1)
void pmac_compose_kernel(const float* __restrict__ features,
                         const unsigned char* __restrict__ mask,
                         const float* __restrict__ gw1, const float* __restrict__ gb1,
                         const float* __restrict__ gln1_g, const float* __restrict__ gln1_b,
                         const float* __restrict__ gw2, const float* __restrict__ gb2,
                         const float* __restrict__ gln2_g, const float* __restrict__ gln2_b,
                         const float* __restrict__ gw3, const float* __restrict__ gb3,
                         const float* __restrict__ wq, const float* __restrict__ bq,
                         const float* __restrict__ wk, const float* __restrict__ bk,
                         const float* __restrict__ wvw, const float* __restrict__ bv,
                         const float* __restrict__ ig, const float* __restrict__ ib,
                         const float* __restrict__ og, const float* __restrict__ ob,
                         float* __restrict__ out) {
  extern __shared__ float smem[];
  float* comp  = smem;                       // [16][768] running composed state
  float* f2bas = comp + ROWS * Dsz;          // [2][16][768] aspect ping/pong
  float* gf    = f2bas + 2 * ROWS * Dsz;     // [16][768] gated fusion (+attn)
  float* qb    = gf + ROWS * Dsz;            // [16][768] q, then softmax-mix
  float* h1    = qb + ROWS * Dsz;            // [16][128]
  float* h2    = h1 + ROWS * Hsz;            // [16][64]
  float* redS  = h2 + ROWS * H2sz;           // [256]
  float* redQ  = redS + 256;                 // [256]
  float* meanS = redQ + 256;                 // [16]
  float* rstdS = meanS + 16;                 // [16]
  float* s0S   = rstdS + 16;                 // [16]
  float* s1S   = s0S + 16;                   // [16]
  float* w0S   = s1S + 16;                   // [16]
  float* w1S   = w0S + 16;                   // [16]
  int*   fiS   = (int*)(w1S + 16);           // [16] first-valid aspect
  int*   fuS   = fiS + 16;                   // [16] fuse flag

  const int tid   = threadIdx.x;
  const int lane  = tid & 31;
  const int wav   = tid >> 5;                // wave id 0..7
  const int m16   = lane & 15;               // A-row / B-col within tile
  const int kh    = lane >> 4;               // K-half select
  const int rbase = kh * 8;                  // C/D row base for this lane
  const int b0    = blockIdx.x * ROWS;

  // Kick off async prefetch of aspect 0 (step 0 uses buffer 0).
  prefetch_f2(features, b0, 0, f2bas, tid);

  // ---- seed: first valid aspect per row (argmax over bool = first True, else 0)
  if (tid < ROWS) {
    int bb = b0 + tid;
    int fi = 0;
    for (int a = 0; a < Asz; ++a)
      if (mask[bb * Asz + a]) { fi = a; break; }
    fiS[tid] = fi;
  }
  __syncthreads();
  for (int idx = tid; idx < ROWS * Dsz; idx += TPB) {
    int r = idx / Dsz, c = idx % Dsz;
    comp[idx] = features[((size_t)(b0 + r) * Asz + fiS[r]) * Dsz + c];
  }

  const float inv_scale = rsqrtf((float)Dsz);

  for (int l = 0; l < Lsz; ++l) {
    const float* gw1_l = gw1 + (size_t)l * 2 * Dsz * Hsz;
    const float* gb1_l = gb1 + l * Hsz;
    const float* g1g = gln1_g + l * Hsz, * g1b = gln1_b + l * Hsz;
    const float* gw2_l = gw2 + (size_t)l * Hsz * H2sz;
    const float* gb2_l = gb2 + l * H2sz;
    const float* g2g = gln2_g + l * H2sz, * g2b = gln2_b + l * H2sz;
    const float* gw3_l = gw3 + (size_t)l * H2sz * Dsz;
    const float* gb3_l = gb3 + l * Dsz;
    const float* wq_l = wq + (size_t)l * Dsz * Dsz, * bq_l = bq + l * Dsz;
    const float* wk_l = wk + (size_t)l * Dsz * Dsz, * bk_l = bk + l * Dsz;
    const float* wv_l = wvw + (size_t)l * Dsz * Dsz, * bv_l = bv + l * Dsz;
    const float* ig_l = ig + l * Dsz, * ib_l = ib + l * Dsz;
    const float* og_l = og + l * Dsz, * ob_l = ob + l * Dsz;

    for (int a = 0; a < Asz; ++a) {
      const int step = l * Asz + a;
      // Offset arithmetic on one LDS base (NOT a pointer select) so address-
      // space inference keeps every f2 access on the DS path.
      const float* f2v = f2bas + (step & 1) * (ROWS * Dsz);

      // Retire this step's async prefetch (issued during the previous step),
      // publish f2v + fuse flags to the whole workgroup.
      if (tid < ROWS) {
        int bb = b0 + tid;
        fuS[tid] = (mask[bb * Asz + a] != 0) && (a != fiS[tid]);
      }
      wait_async_all();
      __syncthreads();

      // Overlap the NEXT aspect's HBM->LDS transfer with this step's compute.
      {
        int sn = step + 1;
        if (sn < Lsz * Asz)
          prefetch_f2(features, b0, sn & (Asz - 1),
                      f2bas + (sn & 1) * (ROWS * Dsz), tid);
      }

      bool any = false;
      for (int r = 0; r < ROWS; ++r) any = any || (fuS[r] != 0);
      if (!any) continue;  // block-uniform skip: step is identity for all 16 rows

      // ---- gate MLP 1: h1 = [comp|f2] @ gw1 + gb1 ; LN ; GELU
      {
        int n0 = wav * 16;  // 8 tiles == 8 waves
        v8f acc = gemm_cat(comp, f2v, gw1_l, Hsz, n0, m16, kh);
        store_tile(h1, Hsz, acc, n0, m16, rbase, gb1_l);
      }
      __syncthreads();
      row_ln_act(h1, Hsz, g1g, g1b, redS, redQ, meanS, rstdS, tid, 1);

      // ---- gate MLP 2: h2 = h1 @ gw2 + gb2 ; LN ; GELU
      if (wav < 4) {
        int n0 = wav * 16;
        v8f acc = gemm_tile(h1, Hsz, gw2_l, H2sz, Hsz, n0, m16, kh);
        store_tile(h2, H2sz, acc, n0, m16, rbase, gb2_l);
      }
      __syncthreads();
      row_ln_act(h2, H2sz, g2g, g2b, redS, redQ, meanS, rstdS, tid, 1);

      // ---- gate = sigmoid(h2 @ gw3 + gb3); gf = g*comp + (1-g)*f2
      for (int t = wav; t < Dsz / 16; t += 8) {
        int n0 = t * 16;
        v8f acc = gemm_tile(h2, H2sz, gw3_l, Dsz, H2sz, n0, m16, kh);
        int col = n0 + m16;
        float gb3v = gb3_l[col];
#pragma unroll
        for (int r = 0; r < 8; ++r) {
          int row = rbase + r;
          float g = 1.0f / (1.0f + expf(-(acc[r] + gb3v)));
          gf[row * Dsz + col] =
              g * comp[row * Dsz + col] + (1.0f - g) * f2v[row * Dsz + col];
        }
      }
      __syncthreads();

      // ---- q = gf @ wq + bq
      for (int t = wav; t < Dsz / 16; t += 8) {
        int n0 = t * 16;
        v8f acc = gemm_tile(gf, Dsz, wq_l, Dsz, Dsz, n0, m16, kh);
        store_tile(qb, Dsz, acc, n0, m16, rbase, bq_l);
      }
      if (tid < ROWS) { s0S[tid] = 0.0f; s1S[tid] = 0.0f; }
      __syncthreads();

      // ---- scores: stream k_comp / k_f2 tiles (shared wk B-operand), dot with q
      for (int t = wav; t < Dsz / 16; t += 8) {
        int n0 = t * 16;
        v8f ac, af;
        gemm_tile_dual(comp, f2v, Dsz, wk_l, Dsz, Dsz, n0, m16, kh, ac, af);
        int col = n0 + m16;
        float bkv = bk_l[col];
#pragma unroll
        for (int r = 0; r < 8; ++r) {
          int row = rbase + r;
          float qv = qb[row * Dsz + col];
          atomicAdd(&s0S[row], (ac[r] + bkv) * qv);  // ds_add_f32
          atomicAdd(&s1S[row], (af[r] + bkv) * qv);
        }
      }
      __syncthreads();
      if (tid < ROWS) {
        float x0 = s0S[tid] * inv_scale, x1 = s1S[tid] * inv_scale;
        float mx = fmaxf(x0, x1);
        float e0 = expf(x0 - mx), e1 = expf(x1 - mx);
        float inv = 1.0f / (e0 + e1);
        w0S[tid] = e0 * inv;
        w1S[tid] = e1 * inv;
      }
      __syncthreads();

      // ---- attn = (w0*comp + w1*f2) @ wv + bv   (w0+w1==1 folds the two V GEMMs)
      for (int idx = tid; idx < ROWS * Dsz; idx += TPB) {
        int r = idx / Dsz;
        qb[idx] = w0S[r] * comp[idx] + w1S[r] * f2v[idx];
      }
      __syncthreads();
      for (int t = wav; t < Dsz / 16; t += 8) {
        int n0 = t * 16;
        v8f acc = gemm_tile(qb, Dsz, wv_l, Dsz, Dsz, n0, m16, kh);
        int col = n0 + m16;
        float bvv = bv_l[col];
#pragma unroll
        for (int r = 0; r < 8; ++r) {
          int row = rbase + r;
          gf[row * Dsz + col] += acc[r] + bvv;  // gf + attn
        }
      }
      __syncthreads();

      // ---- LN(ig,ib) then LN(og,ob) with masked writeback into comp
      row_ln_act(gf, Dsz, ig_l, ib_l, redS, redQ, meanS, rstdS, tid, 0);
      {
        int row = tid >> 4, slot = tid & 15;
        float s = 0.f, q = 0.f;
        for (int c = slot; c < Dsz; c += 16) {
          float v = gf[row * Dsz + c];
          s += v;
          q += v * v;
        }
        redS[row * 16 + slot] = s;
        redQ[row * 16 + slot] = q;
        __syncthreads();
        if (tid < 16) {
          float ts = 0.f, tq = 0.f;
          for (int i = 0; i < 16; ++i) { ts += redS[tid * 16 + i]; tq += redQ[tid * 16 + i]; }
          float mean = ts / (float)Dsz;
          float var = tq / (float)Dsz - mean * mean;
          meanS[tid] = mean;
          rstdS[tid] = rsqrtf(var + 1e-5f);
        }
        __syncthreads();
        float mean = meanS[row], rr = rstdS[row];
        int fm = fuS[row];
        for (int c = slot; c < Dsz; c += 16) {
          float v = (gf[row * Dsz + c] - mean) * rr * og_l[c] + ob_l[c];
          if (fm) comp[row * Dsz + c] = v;
        }
        __syncthreads();
      }
    }
  }

  // ---- broadcast composed to valid aspect slots, zeros elsewhere
  for (int idx = tid; idx < ROWS * Asz * Dsz; idx += TPB) {
    int r = idx / (Asz * Dsz);
    int rem = idx % (Asz * Dsz);
    int a = rem / Dsz, c = rem % Dsz;
    float v = mask[(size_t)(b0 + r) * Asz + a] ? comp[r * Dsz + c] : 0.0f;
    out[((size_t)(b0 + r) * Asz + a) * Dsz + c] = v;
  }
}

extern "C" void kernel_launch(void* const* d_in, const int* in_sizes, int n_in,
                              void* d_out, int out_size, void* d_ws, size_t ws_size,
                              hipStream_t stream) {
  (void)in_sizes; (void)n_in; (void)out_size; (void)d_ws; (void)ws_size;
  const float* features = (const float*)d_in[0];
  const unsigned char* mask = (const unsigned char*)d_in[1];  // numpy bool8
  const float* gw1 = (const float*)d_in[2];
  const float* gb1 = (const float*)d_in[3];
  const float* gln1_g = (const float*)d_in[4];
  const float* gln1_b = (const float*)d_in[5];
  const float* gw2 = (const float*)d_in[6];
  const float* gb2 = (const float*)d_in[7];
  const float* gln2_g = (const float*)d_in[8];
  const float* gln2_b = (const float*)d_in[9];
  const float* gw3 = (const float*)d_in[10];
  const float* gb3 = (const float*)d_in[11];
  const float* wq = (const float*)d_in[12];
  const float* bq = (const float*)d_in[13];
  const float* wk = (const float*)d_in[14];
  const float* bk = (const float*)d_in[15];
  const float* wvw = (const float*)d_in[16];
  const float* bv = (const float*)d_in[17];
  const float* ig = (const float*)d_in[18];
  const float* ib = (const float*)d_in[19];
  const float* og = (const float*)d_in[20];
  const float* ob = (const float*)d_in[21];
  float* out = (float*)d_out;

  // Allow ~255KB dynamic LDS (WGP has 320KB). Host-side attribute set is not a
  // stream op, safe under graph capture; idempotent and deterministic.
  (void)hipFuncSetAttribute((const void*)pmac_compose_kernel,
                            hipFuncAttributeMaxDynamicSharedMemorySize,
                            (int)SMEM_BYTES);

  dim3 grid(Bsz / ROWS), block(TPB);
  pmac_compose_kernel<<<grid, block, SMEM_BYTES, stream>>>(
      features, mask, gw1, gb1, gln1_g, gln1_b, gw2, gb2, gln2_g, gln2_b,
      gw3, gb3, wq, bq, wk, bk, wvw, bv, ig, ib, og, ob, out);
}